// PortCoupling_36129264894531
// MI455X (gfx1250) — compile-verified
//
#include <hip/hip_runtime.h>
#include <stdint.h>

// Problem constants (match reference)
#define BATCH     8192
#define DIM_Q     1024
#define DIM_U     1024
#define NCHARTS   16
#define TOPK      2
#define MAXTILES  (BATCH / 16)   // worst case: every row routes to one chart
#define NB        512            // output columns per workgroup
#define KB        16             // k-chunk staged in LDS per iteration
#define STR       20             // padded LDS row stride in dwords (mult. of 4, conflict-free)

typedef __attribute__((ext_vector_type(2))) float v2f;
typedef __attribute__((ext_vector_type(8))) float v8f;
typedef __attribute__((ext_vector_type(4))) int   v4i;
typedef __attribute__((ext_vector_type(4))) unsigned int u32x4;
typedef __attribute__((ext_vector_type(4))) int   i32x4;
typedef __attribute__((ext_vector_type(8))) int   i32x8;

// ---- CDNA5 feature detection (device pass only; host pass sees fallbacks) ----
#if defined(__AMDGCN__) && __has_builtin(__builtin_amdgcn_tensor_load_to_lds) && \
    __has_builtin(__builtin_amdgcn_s_wait_tensorcnt)
#define HAVE_TDM 1
#else
#define HAVE_TDM 0
#endif
#if defined(__AMDGCN__) && __has_builtin(__builtin_amdgcn_global_load_async_to_lds_b128)
#define HAVE_ASYNC 1
#else
#define HAVE_ASYNC 0
#endif

// Async-to-LDS builtin takes v4i pointers in AS1 (global) / AS3 (LDS).
#define AS1V4(p) ((__attribute__((address_space(1))) v4i*)(p))
#define AS3V4(p) ((__attribute__((address_space(3))) v4i*)(p))
#define AS3RAW(p) ((__attribute__((address_space(3))) void*)(p))

__device__ __forceinline__ void wait_async_lds() {
#if HAVE_ASYNC
#if __has_builtin(__builtin_amdgcn_s_wait_asynccnt)
  __builtin_amdgcn_s_wait_asynccnt(0);
#else
  asm volatile("s_wait_asynccnt 0x0" ::: "memory");
#endif
#endif
}

// ---------------------------------------------------------------------------
// Pass 0a: zero the per-(slot,chart) counters (workspace persists across
// graph replays, so re-zero every call).
// ---------------------------------------------------------------------------
__global__ void zero_counts_kernel(int* __restrict__ counts) {
  if (threadIdx.x < TOPK * NCHARTS) counts[threadIdx.x] = 0;
}

// ---------------------------------------------------------------------------
// Pass 0b: top-2 routing. One thread per row; first-occurrence tie-break
// matches jax.lax.top_k; gates normalized with the 1e-8 clamp. List order is
// nondeterministic but FP-irrelevant (each row's dots use a fixed k-order).
// ---------------------------------------------------------------------------
__global__ void route_topk_kernel(const float* __restrict__ wts,
                                  int*   __restrict__ counts,
                                  int*   __restrict__ rows,
                                  float* __restrict__ gates) {
  int r = blockIdx.x * blockDim.x + threadIdx.x;
  if (r >= BATCH) return;

  float w[NCHARTS];
  const float4* wp = (const float4*)(wts + (size_t)r * NCHARTS);
#pragma unroll
  for (int i = 0; i < NCHARTS / 4; ++i) {
    float4 v = wp[i];
    w[4 * i + 0] = v.x; w[4 * i + 1] = v.y;
    w[4 * i + 2] = v.z; w[4 * i + 3] = v.w;
  }

  int i0 = 0; float v0 = w[0];
#pragma unroll
  for (int i = 1; i < NCHARTS; ++i) if (w[i] > v0) { v0 = w[i]; i0 = i; }
  int i1 = (i0 == 0) ? 1 : 0; float v1 = -3.4e38f;
#pragma unroll
  for (int i = 0; i < NCHARTS; ++i) if (i != i0 && w[i] > v1) { v1 = w[i]; i1 = i; }

  float s  = fmaxf(v0 + v1, 1e-8f);
  float g0 = v0 / s, g1 = v1 / s;

  int p0 = atomicAdd(&counts[i0], 1);
  rows [(size_t)i0 * BATCH + p0] = r;
  gates[(size_t)i0 * BATCH + p0] = g0;

  int p1 = atomicAdd(&counts[NCHARTS + i1], 1);
  rows [(size_t)(NCHARTS + i1) * BATCH + p1] = r;
  gates[(size_t)(NCHARTS + i1) * BATCH + p1] = g1;
}

// ---------------------------------------------------------------------------
// Grouped-GEMM pass for one routing slot.
//   out[row, n] (SLOT==0: =, SLOT==1: +=) gate * sum_k q[row,k] * W[c][n,k]
// Workgroup = (chart, 16-row tile, 512-column block); 256 threads = 8 waves,
// each wave owns 4 WMMA 16x16 column tiles.
// Double-buffered LDS pipeline: the 512x16 W tile is fetched by the Tensor
// Data Mover (D# with pad_enable reproducing the stride-20 padded layout);
// the 16 gathered q rows use GLOBAL_LOAD_ASYNC_TO_LDS_B128. Next chunk is in
// flight while WMMAs consume the current one; S_WAIT_TENSORCNT/ASYNCCNT 0
// precede the workgroup barrier.
// ---------------------------------------------------------------------------
template <int SLOT>
__global__ __launch_bounds__(256) void moe_gemm_kernel(
    const float* __restrict__ q,      const float* __restrict__ Wstack,
    const int*   __restrict__ counts, const int*   __restrict__ rows,
    const float* __restrict__ gates,  float* __restrict__ out) {
  __shared__ float sW[2][NB * STR];     // 2 x (512 x 16 padded)  = 80 KB
  __shared__ float sQ[2][16 * STR];     // 2 x (16 x 16 padded)
  __shared__ int   rowT[16];
  __shared__ float gateT[16];

  const int chart = blockIdx.x >> 9;            // / MAXTILES
  const int tile  = blockIdx.x & (MAXTILES - 1);
  const int nbase = blockIdx.y * NB;
  const int count = counts[SLOT * NCHARTS + chart];
  if (tile * 16 >= count) return;               // uniform early exit

  const int tid  = threadIdx.x;
  const int lane = tid & 31;
  const int wave = tid >> 5;
  const int half = lane >> 4;                   // selects K pair {0,1}/{2,3}
  const int lm   = lane & 15;

  if (tid < 16) {
    int mg   = tile * 16 + tid;
    int safe = (mg < count) ? mg : count - 1;
    size_t base = (size_t)(SLOT * NCHARTS + chart) * BATCH + safe;
    rowT[tid]  = rows[base];
    gateT[tid] = (mg < count) ? gates[base] : 0.0f;
  }
  __syncthreads();

  const float* Wc = Wstack + (size_t)chart * DIM_U * DIM_Q;

  v8f acc[4];
#pragma unroll
  for (int i = 0; i < 4; ++i) acc[i] = 0.0f;

  const int c4 = (tid & 3) * 4;                 // staging column (float4)
  const int r0 = tid >> 2;                      // staging row 0..63

  // Issue the staging traffic for one (buffer, k-chunk).
  auto issue_stage = [&](int buf, int kb) {
    // ---- W tile: 512 rows x 16 k, uniform -> Tensor Data Mover ----
#if HAVE_TDM
    if (tid == 0) {
      uint64_t ga  = (uint64_t)(uintptr_t)(Wc + (size_t)nbase * DIM_Q + kb);
      uint32_t lds = (uint32_t)(uintptr_t)AS3RAW(&sW[buf][0]);
      u32x4 g0 = {};
      g0[0] = 1u;                                // count=1, user descriptor
      g0[1] = lds;                               // lds_addr (bytes)
      g0[2] = (uint32_t)ga;                      // global_addr[31:0]
      g0[3] = ((uint32_t)(ga >> 32) & 0x01FFFFFFu) | (2u << 30);  // addr[56:32] | type=2
      i32x8 g1 = {};
      // data_size=2 (4B) | pad_enable | pad_interval=3 (16 dw) | pad_amount=3 (4 dw)
      g1[0] = (2 << 16) | (1 << 20) | (3 << 22) | (3 << 25);
      g1[1] = (DIM_Q & 0xFFFF) << 16;            // tensor_dim0[15:0]  (bits 63:48)
      g1[2] = (DIM_Q >> 16) | ((DIM_U & 0xFFFF) << 16);  // dim0 hi | tensor_dim1 lo
      g1[3] = (DIM_U >> 16) | (KB << 16);        // tensor_dim1 hi | tile_dim0=16
      g1[4] = NB;                                // tile_dim1=512 | tile_dim2=0
      g1[5] = DIM_Q;                             // tensor_dim0_stride[31:0] = 1024
      g1[6] = 0;                                 // stride hi | tensor_dim1_stride lo
      g1[7] = 0;
      i32x4 gz = {};
#if __clang_major__ >= 23
      i32x8 gz8 = {};
      __builtin_amdgcn_tensor_load_to_lds(g0, g1, gz, gz, gz8, 0);
#else
      __builtin_amdgcn_tensor_load_to_lds(g0, g1, gz, gz, 0);
#endif
    }
#elif HAVE_ASYNC
#pragma unroll
    for (int p = 0; p < NB / 64; ++p) {
      int rr = p * 64 + r0;
      __builtin_amdgcn_global_load_async_to_lds_b128(
          AS1V4(Wc + (size_t)(nbase + rr) * DIM_Q + kb + c4),
          AS3V4(&sW[buf][rr * STR + c4]), 0, 0);
    }
#else
#pragma unroll
    for (int p = 0; p < NB / 64; ++p) {
      int rr = p * 64 + r0;
      float4 v = *(const float4*)(Wc + (size_t)(nbase + rr) * DIM_Q + kb + c4);
      *(float4*)(&sW[buf][rr * STR + c4]) = v;
    }
#endif
    // ---- q tile: 16 gathered rows x 16 k (row-indexed, so async not TDM) ----
#if HAVE_ASYNC
    if (tid < 64)
      __builtin_amdgcn_global_load_async_to_lds_b128(
          AS1V4(q + (size_t)rowT[r0] * DIM_Q + kb + c4),
          AS3V4(&sQ[buf][r0 * STR + c4]), 0, 0);
#else
    if (tid < 64) {
      float4 v = *(const float4*)(q + (size_t)rowT[r0] * DIM_Q + kb + c4);
      *(float4*)(&sQ[buf][r0 * STR + c4]) = v;
    }
#endif
  };

  auto wait_stage = [&]() {
#if HAVE_TDM
    __builtin_amdgcn_s_wait_tensorcnt(0);
#endif
    wait_async_lds();
  };

  // ---- Software pipeline: fetch chunk k+1 while WMMAs consume chunk k ----
  issue_stage(0, 0);
  wait_stage();
  __syncthreads();

  int buf = 0;
  for (int kb = 0; kb < DIM_Q; kb += KB) {
    const bool has_next = (kb + KB) < DIM_Q;
    if (has_next) issue_stage(buf ^ 1, kb + KB);

    const float* sQb = &sQ[buf][0];
    const float* sWb = &sW[buf][0];
#pragma unroll
    for (int kt = 0; kt < KB; kt += 4) {
      // A fragment: 16x4 f32, lane=M, vgpr pair=K (lane-half selects K 0,1/2,3)
      v2f a = *(const v2f*)(sQb + lm * STR + kt + 2 * half);
#pragma unroll
      for (int nt = 0; nt < 4; ++nt) {
        // B fragment: 4x16 f32, lane=N, B[k][n] = W[nbase+n][k]
        v2f b = *(const v2f*)(sWb + (wave * 64 + nt * 16 + lm) * STR + kt + 2 * half);
        acc[nt] = __builtin_amdgcn_wmma_f32_16x16x4_f32(
            false, a, false, b, (short)0, acc[nt], false, false);
      }
    }

    if (has_next) wait_stage();
    __syncthreads();
    buf ^= 1;
  }

  // Epilogue: scale by gate, scatter to gathered rows.
  // SLOT 0 covers every row exactly once -> plain store (no pre-zero needed).
  // SLOT 1 read-add-store; ordered after SLOT 0 by stream dependency.
#pragma unroll
  for (int nt = 0; nt < 4; ++nt) {
    const int col = nbase + wave * 64 + nt * 16 + lm;
#pragma unroll
    for (int rI = 0; rI < 8; ++rI) {
      const int M = rI + 8 * half;              // C layout: M = vgpr + 8*half
      if (tile * 16 + M < count) {
        float v  = acc[nt][rI] * gateT[M];
        float* p = out + (size_t)rowT[M] * DIM_U + col;
        if (SLOT == 0) *p = v;
        else           *p = *p + v;
      }
    }
  }
}

// ---------------------------------------------------------------------------
extern "C" void kernel_launch(void* const* d_in, const int* in_sizes, int n_in,
                              void* d_out, int out_size, void* d_ws, size_t ws_size,
                              hipStream_t stream) {
  (void)in_sizes; (void)n_in; (void)out_size; (void)ws_size;

  const float* q   = (const float*)d_in[0];   // [B, DIM_Q]
  const float* wts = (const float*)d_in[1];   // [B, NCHARTS]
  const float* Wst = (const float*)d_in[2];   // [NCHARTS, DIM_U, DIM_Q]
  float*       out = (float*)d_out;           // [B, DIM_U]

  // Workspace layout (~2.1 MB): counts | row lists | gate lists
  int*   counts = (int*)d_ws;
  int*   rows   = counts + 64;                            // 256B offset
  float* gates  = (float*)(rows + (size_t)TOPK * NCHARTS * BATCH);

  zero_counts_kernel<<<1, 64, 0, stream>>>(counts);
  route_topk_kernel<<<BATCH / 256, 256, 0, stream>>>(wts, counts, rows, gates);

  dim3 grid(NCHARTS * MAXTILES, DIM_U / NB, 1);
  moe_gemm_kernel<0><<<grid, 256, 0, stream>>>(q, Wst, counts, rows, gates, out);
  moe_gemm_kernel<1><<<grid, 256, 0, stream>>>(q, Wst, counts, rows, gates, out);
}